// StatusGCN_66082366816448
// MI455X (gfx1250) — compile-verified
//
#include <hip/hip_runtime.h>
#include <math.h>

// ---------------------------------------------------------------------------
// Types for WMMA fragments (wave32, gfx1250)
// ---------------------------------------------------------------------------
typedef __attribute__((ext_vector_type(2))) float v2f;
typedef __attribute__((ext_vector_type(8))) float v8f;
typedef int vi4 __attribute__((vector_size(4 * sizeof(int))));  // b128 payload

// ---------------------------------------------------------------------------
// Problem constants (match reference)
// ---------------------------------------------------------------------------
#define NNODES 100000
#define NEDGES 3200000
#define NFEAT  512
#define NHID   256
#define NCLASS 64

// ---------------------------------------------------------------------------
// CDNA5 async global->LDS copy (GLOBAL_LOAD_ASYNC_TO_LDS_B128, ASYNCcnt)
//   builtin signature (from clang diagnostic): (vi4 as1*, vi4 as3*, Iimm, Iimm)
// ---------------------------------------------------------------------------
#if defined(__has_builtin)
#  if __has_builtin(__builtin_amdgcn_global_load_async_to_lds_b128)
#    define HAVE_ASYNC_LDS 1
#  endif
#  if __has_builtin(__builtin_amdgcn_s_wait_asynccnt)
#    define HAVE_WAIT_ASYNC 1
#  endif
#endif
#ifndef HAVE_ASYNC_LDS
#  define HAVE_ASYNC_LDS 0
#endif
#ifndef HAVE_WAIT_ASYNC
#  define HAVE_WAIT_ASYNC 0
#endif

typedef __attribute__((address_space(1))) vi4* g4p_t;  // global int4*
typedef __attribute__((address_space(3))) vi4* l4p_t;  // LDS int4*

__device__ __forceinline__ void async_copy_f4(const float* __restrict__ g, float* l) {
#if HAVE_ASYNC_LDS
    __builtin_amdgcn_global_load_async_to_lds_b128(
        (g4p_t)(uintptr_t)g,   // global flat addr == AS1 addr (bit-identical)
        (l4p_t)l,              // generic -> AS3 addrspacecast
        /*offset=*/0, /*cpol=*/0);
#else
    *(float4*)l = *(const float4*)g;   // synchronous fallback
#endif
}

__device__ __forceinline__ void wait_async0() {
#if HAVE_ASYNC_LDS
#  if HAVE_WAIT_ASYNC
    __builtin_amdgcn_s_wait_asynccnt(0);
#  else
    asm volatile("s_wait_asynccnt 0" ::: "memory");
#  endif
#endif
}

// ---------------------------------------------------------------------------
// WMMA f32 GEMM:  Y[M,N] = X[M,K] @ W[K,N]
//   - block = 128 threads = 4 wave32 waves; each wave computes 16 rows
//   - wave tile = 16 x (16*NTILES) columns, v8f accumulators
//   - W K-slab (KC x NB) double-buffered in LDS via async global->LDS loads
//   - A fragments software-pipelined in registers (LOADcnt hides latency)
//   - uses V_WMMA_F32_16X16X4_F32 (full f32 precision, f32 accumulate)
// Requirements: K % 32 == 0, N % (16*NTILES) == 0, M % 16 == 0.
// ---------------------------------------------------------------------------
template <int NTILES>
__global__ __launch_bounds__(128)
void wmma_gemm_f32(const float* __restrict__ X, const float* __restrict__ W,
                   float* __restrict__ Y, int M, int N, int K) {
    constexpr int KC     = 32;            // K chunk staged in LDS
    constexpr int KSTEPS = KC / 4;        // 8 WMMA k-steps per chunk
    constexpr int NB     = NTILES * 16;   // block column width
    constexpr int F4     = (KC * NB) / 4; // float4s per slab
    constexpr int PERTHR = F4 / 128;      // async loads per thread per chunk

    __shared__ float Bs0[KC * NB];
    __shared__ float Bs1[KC * NB];

    const int lane = threadIdx.x & 31;
    const int wave = threadIdx.x >> 5;

    int row0 = blockIdx.x * 64 + wave * 16;
    const int col0 = blockIdx.y * NB;

    // WMMA requires EXEC all-1s: clamp OOB waves, suppress their stores.
    const bool valid = (row0 + 16) <= M;
    if (!valid) row0 = M - 16;

    const int m    = lane & 15;          // row within tile (A) / col (B,D)
    const int koff = (lane >> 4) * 2;    // f32 layout: lanes>=16 hold K+2,K+3

    v8f acc[NTILES];
#pragma unroll
    for (int t = 0; t < NTILES; ++t) acc[t] = (v8f)(0.0f);

    const float* __restrict__ xrow = X + (size_t)(row0 + m) * K;
    const float* __restrict__ wcol = W + col0;

    // issue async loads for one W slab [k0..k0+KC) x [col0..col0+NB)
    auto issueB = [&](int k0, float* buf) {
#pragma unroll
        for (int q = 0; q < PERTHR; ++q) {
            const int f  = threadIdx.x + 128 * q;   // float4 index in slab
            const int kk = (4 * f) / NB;
            const int nn = (4 * f) - kk * NB;
            async_copy_f4(wcol + (size_t)(k0 + kk) * N + nn, buf + 4 * f);
        }
    };
    // load this wave-lane's A fragments for one K chunk
    auto loadA = [&](int k0, v2f* af) {
#pragma unroll
        for (int f = 0; f < KSTEPS; ++f)
            af[f] = *(const v2f*)(xrow + k0 + 4 * f + koff);
    };

    const int NC = K / KC;
    v2f afc[KSTEPS], afn[KSTEPS];

    issueB(0, Bs0);
    loadA(0, afc);

    for (int c = 0; c < NC; ++c) {
        float* bs = (c & 1) ? Bs1 : Bs0;
        wait_async0();          // my slab loads for chunk c are in LDS
        __syncthreads();        // everyone's are; prev compute done reading

        if (c + 1 < NC) {
            issueB((c + 1) * KC, (c & 1) ? Bs0 : Bs1);  // overlap with compute
            loadA((c + 1) * KC, afn);                   // pipelined A frags
        }

#pragma unroll
        for (int ks = 0; ks < KSTEPS; ++ks) {
            const int kr = ks * 4 + koff;
#pragma unroll
            for (int t = 0; t < NTILES; ++t) {
                v2f b;
                b.x = bs[(kr + 0) * NB + t * 16 + m];
                b.y = bs[(kr + 1) * NB + t * 16 + m];
                acc[t] = __builtin_amdgcn_wmma_f32_16x16x4_f32(
                    false, afc[ks], false, b, (short)0, acc[t], false, false);
            }
        }

        if (c + 1 < NC) {
#pragma unroll
            for (int f = 0; f < KSTEPS; ++f) afc[f] = afn[f];
        }
    }

    if (valid) {
        const int rbase = (lane >> 4) * 8;  // C/D layout: lanes>=16 hold M=r+8
#pragma unroll
        for (int t = 0; t < NTILES; ++t) {
#pragma unroll
            for (int r = 0; r < 8; ++r) {
                Y[(size_t)(row0 + r + rbase) * N + col0 + t * 16 + m] = acc[t][r];
            }
        }
    }
}

// ---------------------------------------------------------------------------
// Edge-parallel SpMM:  out[dst] += w * h[src]   (COO scatter-add)
// One thread handles 4 features of one edge (float4 gather, 4 f32 atomics).
// h rows are L2-resident (h1 = 102 MB < 192 MB L2) so gathers hit L2.
// ---------------------------------------------------------------------------
__global__ __launch_bounds__(256)
void spmm_atomic_kernel(const float* __restrict__ h, const int* __restrict__ src,
                        const int* __restrict__ dst, const float* __restrict__ w,
                        float* __restrict__ out, int E, int D) {
    const int D4 = D >> 2;
    const long long t = (long long)blockIdx.x * blockDim.x + threadIdx.x;
    const long long total = (long long)E * D4;
    if (t >= total) return;
    const int e = (int)(t / D4);
    const int j = ((int)(t - (long long)e * D4)) << 2;

    const int   s  = src[e];
    const int   d  = dst[e];
    const float ww = w[e];

    const float4 v = *(const float4*)(h + (size_t)s * D + j);
    float* o = out + (size_t)d * D + j;
    atomicAdd(o + 0, v.x * ww);
    atomicAdd(o + 1, v.y * ww);
    atomicAdd(o + 2, v.z * ww);
    atomicAdd(o + 3, v.w * ww);
}

// ---------------------------------------------------------------------------
// Elementwise epilogue: h = (h + b[j % D]) with optional ReLU.  D power of 2.
// ---------------------------------------------------------------------------
__global__ __launch_bounds__(256)
void bias_act_kernel(float* __restrict__ h, const float* __restrict__ b,
                     long long total, int D, int do_relu) {
    const long long i = (long long)blockIdx.x * blockDim.x + threadIdx.x;
    if (i >= total) return;
    float v = h[i] + b[(int)i & (D - 1)];
    if (do_relu) v = fmaxf(v, 0.0f);
    h[i] = v;
}

// ---------------------------------------------------------------------------
// Final: out = log_softmax(acc + r) over 64 classes; one wave32 per row.
// ---------------------------------------------------------------------------
__global__ __launch_bounds__(128)
void logsoftmax_kernel(const float* __restrict__ acc, const float* __restrict__ r,
                       float* __restrict__ out, int M) {
    const int lane = threadIdx.x & 31;
    const int wave = threadIdx.x >> 5;
    const int row  = blockIdx.x * 4 + wave;
    if (row >= M) return;

    const float* a = acc + (size_t)row * NCLASS;
    const float v0 = a[lane]      + r[lane];
    const float v1 = a[lane + 32] + r[lane + 32];

    float mx = fmaxf(v0, v1);
#pragma unroll
    for (int off = 16; off > 0; off >>= 1)
        mx = fmaxf(mx, __shfl_xor(mx, off, 32));

    float s = __expf(v0 - mx) + __expf(v1 - mx);
#pragma unroll
    for (int off = 16; off > 0; off >>= 1)
        s += __shfl_xor(s, off, 32);

    const float lse = __logf(s) + mx;
    out[(size_t)row * NCLASS + lane]      = v0 - lse;
    out[(size_t)row * NCLASS + lane + 32] = v1 - lse;
}

// ---------------------------------------------------------------------------
// Host orchestration
// ---------------------------------------------------------------------------
extern "C" void kernel_launch(void* const* d_in, const int* in_sizes, int n_in,
                              void* d_out, int out_size, void* d_ws, size_t ws_size,
                              hipStream_t stream) {
    const float* x   = (const float*)d_in[0];
    const int*   src = (const int*)  d_in[1];
    const int*   dst = (const int*)  d_in[2];
    const float* ew  = (const float*)d_in[3];
    const float* W1  = (const float*)d_in[4];
    const float* b1  = (const float*)d_in[5];
    const float* W2  = (const float*)d_in[6];
    const float* b2  = (const float*)d_in[7];
    const float* W3  = (const float*)d_in[8];
    const float* rr  = (const float*)d_in[9];
    float* out = (float*)d_out;

    // Workspace layout (floats): bufA [N*256], bufB [N*256]
    float* bufA = (float*)d_ws;                         // GEMM outputs (tmp)
    float* bufB = bufA + (size_t)NNODES * NHID;         // SpMM accumulators
    float* bufC = bufB;                                 // layer-2 acc (N*64) aliases bufB
    float* bufD = bufB + (size_t)NNODES * NCLASS;       // layer-3 acc (N*64)

    const int mblocks = (NNODES + 63) / 64;

    // ---- Layer 1: h1 = relu(spmm(x @ W1) + b1) ----
    (void)hipMemsetAsync(bufB, 0, (size_t)NNODES * NHID * sizeof(float), stream);
    {
        dim3 grid(mblocks, NHID / 128);
        wmma_gemm_f32<8><<<grid, 128, 0, stream>>>(x, W1, bufA, NNODES, NHID, NFEAT);
    }
    {
        const long long total = (long long)NEDGES * (NHID / 4);
        const int blocks = (int)((total + 255) / 256);
        spmm_atomic_kernel<<<blocks, 256, 0, stream>>>(bufA, src, dst, ew, bufB, NEDGES, NHID);
    }
    {
        const long long total = (long long)NNODES * NHID;
        const int blocks = (int)((total + 255) / 256);
        bias_act_kernel<<<blocks, 256, 0, stream>>>(bufB, b1, total, NHID, 1);
    }

    // ---- Layer 2: h2 = spmm(h1 @ W2) + b2 ----
    {
        dim3 grid(mblocks, 1);
        wmma_gemm_f32<4><<<grid, 128, 0, stream>>>(bufB, W2, bufA, NNODES, NCLASS, NHID);
    }
    (void)hipMemsetAsync(bufC, 0, (size_t)NNODES * NCLASS * sizeof(float), stream);
    {
        const long long total = (long long)NEDGES * (NCLASS / 4);
        const int blocks = (int)((total + 255) / 256);
        spmm_atomic_kernel<<<blocks, 256, 0, stream>>>(bufA, src, dst, ew, bufC, NEDGES, NCLASS);
    }
    {
        const long long total = (long long)NNODES * NCLASS;
        const int blocks = (int)((total + 255) / 256);
        bias_act_kernel<<<blocks, 256, 0, stream>>>(bufC, b2, total, NCLASS, 0);
    }

    // ---- Layer 3: h3 = spmm(h2 @ W3) + r ; out = log_softmax(h3) ----
    {
        dim3 grid(mblocks, 1);
        wmma_gemm_f32<4><<<grid, 128, 0, stream>>>(bufC, W3, bufA, NNODES, NCLASS, NCLASS);
    }
    (void)hipMemsetAsync(bufD, 0, (size_t)NNODES * NCLASS * sizeof(float), stream);
    {
        const long long total = (long long)NEDGES * (NCLASS / 4);
        const int blocks = (int)((total + 255) / 256);
        spmm_atomic_kernel<<<blocks, 256, 0, stream>>>(bufA, src, dst, ew, bufD, NEDGES, NCLASS);
    }
    {
        const int blocks = (NNODES + 3) / 4;
        logsoftmax_kernel<<<blocks, 128, 0, stream>>>(bufD, rr, out, NNODES);
    }
}